// GraphAttention_Embs_89094801588705
// MI455X (gfx1250) — compile-verified
//
#include <hip/hip_runtime.h>
#include <hip/hip_bf16.h>

// ---------------------------------------------------------------------------
// GATv2 hetero-GNN forward for MI455X (gfx1250, wave32).
//   - lin_l/lin_r GEMMs via V_WMMA_F32_16X16X4_F32, 16x64 strip per wave
//   - edge softmax/aggregation: wave-per-edge, b128 gathers, head-mean
//     folded into the scatter (64 atomics/edge instead of 256)
// ---------------------------------------------------------------------------

#define HEADS 4
#define CDIM  64
#define FDIM  (HEADS * CDIM)   // 256

typedef __attribute__((ext_vector_type(2))) float v2f;
typedef __attribute__((ext_vector_type(8))) float v8f;

// ---------------- helpers ----------------

__device__ __forceinline__ float wave_sum32(float v) {
#pragma unroll
  for (int off = 16; off > 0; off >>= 1) v += __shfl_xor(v, off, 32);
  return v;
}

// order-preserving float <-> uint encoding (for atomic max on signed floats)
__device__ __forceinline__ unsigned f2ord(float f) {
  unsigned b = __float_as_uint(f);
  return (b & 0x80000000u) ? ~b : (b | 0x80000000u);
}
__device__ __forceinline__ float ord2f(unsigned k) {
  unsigned b = (k & 0x80000000u) ? (k & 0x7FFFFFFFu) : ~k;
  return __uint_as_float(b);
}
#define ORD_NEG_INF 0x007FFFFFu   // f2ord(-inf)

// leaky_relu(a+r, 0.2) dotted with att, 4 channels at once
__device__ __forceinline__ float lrelu_dot4(const float4 a, const float4 r,
                                            const float4 t) {
  float e0 = a.x + r.x, e1 = a.y + r.y, e2 = a.z + r.z, e3 = a.w + r.w;
  e0 = e0 > 0.f ? e0 : 0.2f * e0;
  e1 = e1 > 0.f ? e1 : 0.2f * e1;
  e2 = e2 > 0.f ? e2 : 0.2f * e2;
  e3 = e3 > 0.f ? e3 : 0.2f * e3;
  return t.x * e0 + t.y * e1 + t.z * e2 + t.w * e3;
}

// ---------------- GEMM: Y[n][256] = X[idx[n]][64] @ W[256][64]^T + bias -----
//
// One wave per 16x64 output strip: 4 accumulators share one A fragment,
// 64 x V_WMMA_F32_16X16X4_F32 per wave.
// A-frag (16x4 f32, 2 VGPR): lane<16 -> row=lane, K={k0,k0+1};
//                            lane>=16 -> row=lane-16, K={k0+2,k0+3}.
// B-frag (4x16, 2 VGPR):     v0 = B[k0+2*half][N=lane&15], v1 = next K.
// C/D (8 VGPR): element j -> row = 16*mt + j + 8*half, col = (lane&15).
__global__ void gemm_lin_wmma(const float* __restrict__ X,
                              const int* __restrict__ idx,   // row gather (or null)
                              const float* __restrict__ W,   // [256][64]
                              const float* __restrict__ bias,// [256]
                              float* __restrict__ Y,         // [nrows][256]
                              int nrows) {
  const int w    = (blockIdx.x * blockDim.x + threadIdx.x) >> 5;  // strip id
  const int lane = threadIdx.x & 31;
  const int mt = w >> 2;          // 4 strips of 64 cols each
  const int nt = w & 3;
  if (mt * 16 >= nrows) return;   // wave-uniform: EXEC stays all-1s

  const int r    = lane & 15;
  const int half = lane >> 4;     // 0/1

  const int m   = mt * 16 + r;
  const int row = idx ? idx[m] : m;
  const float* __restrict__ xa = X + (size_t)row * CDIM;
  const int nbase = nt * 64 + r;
  const float* __restrict__ wb0 = W + (size_t)(nbase)      * CDIM;
  const float* __restrict__ wb1 = W + (size_t)(nbase + 16) * CDIM;
  const float* __restrict__ wb2 = W + (size_t)(nbase + 32) * CDIM;
  const float* __restrict__ wb3 = W + (size_t)(nbase + 48) * CDIM;

  v8f acc0 = {}, acc1 = {}, acc2 = {}, acc3 = {};
#pragma unroll
  for (int k0 = 0; k0 < CDIM; k0 += 4) {
    const int k = k0 + 2 * half;
    v2f a, b0, b1, b2, b3;
    a.x  = xa[k];   a.y  = xa[k + 1];
    b0.x = wb0[k];  b0.y = wb0[k + 1];
    b1.x = wb1[k];  b1.y = wb1[k + 1];
    b2.x = wb2[k];  b2.y = wb2[k + 1];
    b3.x = wb3[k];  b3.y = wb3[k + 1];
    acc0 = __builtin_amdgcn_wmma_f32_16x16x4_f32(false, a, false, b0,
                                                 (short)0, acc0, false, false);
    acc1 = __builtin_amdgcn_wmma_f32_16x16x4_f32(false, a, false, b1,
                                                 (short)0, acc1, false, false);
    acc2 = __builtin_amdgcn_wmma_f32_16x16x4_f32(false, a, false, b2,
                                                 (short)0, acc2, false, false);
    acc3 = __builtin_amdgcn_wmma_f32_16x16x4_f32(false, a, false, b3,
                                                 (short)0, acc3, false, false);
  }

  const float bv0 = bias[nbase];
  const float bv1 = bias[nbase + 16];
  const float bv2 = bias[nbase + 32];
  const float bv3 = bias[nbase + 48];
#pragma unroll
  for (int j = 0; j < 8; ++j) {
    const int mrow = mt * 16 + j + 8 * half;
    float* __restrict__ yp = Y + (size_t)mrow * FDIM + nbase;
    yp[0]  = acc0[j] + bv0;
    yp[16] = acc1[j] + bv1;
    yp[32] = acc2[j] + bv2;
    yp[48] = acc3[j] + bv3;
  }
}

// ---------------- softmax-state / output init ----------------
__global__ void init_mkey(unsigned* __restrict__ mkey, int n) {
  int t = blockIdx.x * blockDim.x + threadIdx.x;
  if (t < n) mkey[t] = ORD_NEG_INF;
}

__global__ void init_bias(float* __restrict__ out,
                          const float* __restrict__ bias, int n64) {
  int t = blockIdx.x * blockDim.x + threadIdx.x;
  if (t < n64) out[t] = bias[t & (CDIM - 1)];
}

// ---------------- pass A: per-edge logits + segment max ----------------
// wave per edge; lane owns 4 contiguous channels (b128 gathers).
// First b128 covers heads {0,1} (lane halves), second covers heads {2,3}.
__global__ void edge_logits_max(const float* __restrict__ xl,
                                const float* __restrict__ xr,
                                const int* __restrict__ src,
                                const int* __restrict__ dst,
                                const float* __restrict__ att,   // [4*64]
                                float* __restrict__ lg,          // [E][4]
                                unsigned* __restrict__ mkey,     // [Nd][4]
                                int E) {
  const int w = (blockIdx.x * blockDim.x + threadIdx.x) >> 5;
  if (w >= E) return;
  const int lane = threadIdx.x & 31;
  const int s = src[w];
  const int d = dst[w];
  const float4* __restrict__ pl = (const float4*)(xl + (size_t)s * FDIM);
  const float4* __restrict__ pr = (const float4*)(xr + (size_t)d * FDIM);
  const float4* __restrict__ at = (const float4*)att;

  // lanes 0-15: heads 0 (pA) and 2 (pB); lanes 16-31: heads 1 and 3
  float pA = lrelu_dot4(pl[lane],      pr[lane],      at[lane]);
  float pB = lrelu_dot4(pl[lane + 32], pr[lane + 32], at[lane + 32]);
#pragma unroll
  for (int off = 8; off > 0; off >>= 1) {
    pA += __shfl_xor(pA, off, 32);
    pB += __shfl_xor(pB, off, 32);
  }
  if ((lane & 15) == 0) {
    const int h = lane >> 4;                 // 0 or 1
    lg[(size_t)w * HEADS + h]     = pA;
    lg[(size_t)w * HEADS + h + 2] = pB;
    atomicMax(&mkey[(size_t)d * HEADS + h],     f2ord(pA));
    atomicMax(&mkey[(size_t)d * HEADS + h + 2], f2ord(pB));
  }
}

// ---------------- pass B: exp(logit - max) + segment sum ----------------
__global__ void edge_exp_den(const int* __restrict__ dst,
                             const unsigned* __restrict__ mkey,
                             float* __restrict__ lg,    // in: logits, out: exp
                             float* __restrict__ den,   // [Nd][4]
                             int E4) {
  const int t = blockIdx.x * blockDim.x + threadIdx.x;
  if (t >= E4) return;
  const int e = t >> 2;
  const int h = t & 3;
  const int d = dst[e];
  float m = ord2f(mkey[(size_t)d * HEADS + h]);
  if (!__builtin_isfinite(m)) m = 0.f;   // empty segments -> 0 (matches ref)
  const float ex = __expf(lg[t] - m);
  lg[t] = ex;
  atomicAdd(&den[(size_t)d * HEADS + h], ex);
}

// ---------------- pass C: head-mean-folded weighted scatter ----------
// out[d][c] += 0.25 * sum_h alpha[e,h] * xl[s][h*64+c]; 64 atomics per edge.
__global__ void edge_scatter(const float* __restrict__ xl,
                             const int* __restrict__ src,
                             const int* __restrict__ dst,
                             const float* __restrict__ ex,   // [E][4]
                             const float* __restrict__ den,  // [Nd][4]
                             float* __restrict__ out,        // [Nd][64], bias-seeded
                             int E) {
  const int w = (blockIdx.x * blockDim.x + threadIdx.x) >> 5;
  if (w >= E) return;
  const int lane = threadIdx.x & 31;
  const int s = src[w];
  const int d = dst[w];

  const float4 exv = *(const float4*)(ex  + (size_t)w * HEADS);
  const float4 dnv = *(const float4*)(den + (size_t)d * HEADS);
  const float a0 = 0.25f * exv.x / (dnv.x + 1e-16f);
  const float a1 = 0.25f * exv.y / (dnv.y + 1e-16f);
  const float a2 = 0.25f * exv.z / (dnv.z + 1e-16f);
  const float a3 = 0.25f * exv.w / (dnv.w + 1e-16f);

  const float4* __restrict__ pl = (const float4*)(xl + (size_t)s * FDIM);
  const float4 u = pl[lane];        // head 0 (lanes<16) / head 1
  const float4 v = pl[lane + 32];   // head 2 (lanes<16) / head 3
  const bool lo = lane < 16;
  const float aa = lo ? a0 : a1;
  const float ab = lo ? a2 : a3;
  float p0 = u.x * aa + v.x * ab;
  float p1 = u.y * aa + v.y * ab;
  float p2 = u.z * aa + v.z * ab;
  float p3 = u.w * aa + v.w * ab;
  // combine the two head-halves: channel c = 4*(lane&15)+i
  p0 += __shfl_xor(p0, 16, 32);
  p1 += __shfl_xor(p1, 16, 32);
  p2 += __shfl_xor(p2, 16, 32);
  p3 += __shfl_xor(p3, 16, 32);

  float* __restrict__ po = out + (size_t)d * CDIM + 4 * (lane & 15);
  if (lo) {
    atomicAdd(po + 0, p0);
    atomicAdd(po + 1, p1);
  } else {
    atomicAdd(po + 2, p2);
    atomicAdd(po + 3, p3);
  }
}

// ---------------- relu + layernorm (in place), wave per row ----------------
__global__ void relu_layernorm(float* __restrict__ x,          // [n][64]
                               const float* __restrict__ w,
                               const float* __restrict__ b,
                               int n) {
  const int row = (blockIdx.x * blockDim.x + threadIdx.x) >> 5;
  if (row >= n) return;
  const int lane = threadIdx.x & 31;
  float* __restrict__ p = x + (size_t)row * CDIM;
  float v0 = p[lane];
  float v1 = p[lane + 32];
  v0 = v0 > 0.f ? v0 : 0.f;
  v1 = v1 > 0.f ? v1 : 0.f;
  const float mu = wave_sum32(v0 + v1) * (1.0f / CDIM);
  const float d0 = v0 - mu, d1 = v1 - mu;
  const float var = wave_sum32(d0 * d0 + d1 * d1) * (1.0f / CDIM);
  const float inv = rsqrtf(var + 1e-5f);
  p[lane]      = d0 * inv * w[lane]      + b[lane];
  p[lane + 32] = d1 * inv * w[lane + 32] + b[lane + 32];
}

// ---------------- classifier: edge-wise dot product ----------------
__global__ void edge_dot(const float* __restrict__ os,   // [n][64]
                         const float* __restrict__ ot,   // [n][64]
                         const int* __restrict__ ia,
                         const int* __restrict__ ib,
                         float* __restrict__ pred, int n) {
  const int w = (blockIdx.x * blockDim.x + threadIdx.x) >> 5;
  if (w >= n) return;
  const int lane = threadIdx.x & 31;
  const float* __restrict__ pa = os + (size_t)ia[w] * CDIM;
  const float* __restrict__ pb = ot + (size_t)ib[w] * CDIM;
  const float p =
      wave_sum32(pa[lane] * pb[lane] + pa[lane + 32] * pb[lane + 32]);
  if (lane == 0) pred[w] = p;
}

// ---------------- host-side conv driver ----------------
static void run_gatv2(const float* xsrc, const int* idx_src,
                      const float* xdst, const int* idx_dst,
                      const int* ei, int E,
                      const float* Wl, const float* bl,
                      const float* Wr, const float* br,
                      const float* att, const float* bias,
                      int n_src, int n_dst,
                      float* XL, float* XR, float* EX,
                      float* DEN, unsigned* MKEY,
                      float* out, hipStream_t stream) {
  const int* src = ei;
  const int* dst = ei + E;

  // 4 waves (one 16x64 strip each) per 16-row block
  const int gw_src = ((n_src + 15) / 16) * 4 * 32;
  const int gw_dst = ((n_dst + 15) / 16) * 4 * 32;
  gemm_lin_wmma<<<(gw_src + 255) / 256, 256, 0, stream>>>(
      xsrc, idx_src, Wl, bl, XL, n_src);
  gemm_lin_wmma<<<(gw_dst + 255) / 256, 256, 0, stream>>>(
      xdst, idx_dst, Wr, br, XR, n_dst);

  hipMemsetAsync(DEN, 0, (size_t)n_dst * HEADS * sizeof(float), stream);
  init_mkey<<<(n_dst * HEADS + 255) / 256, 256, 0, stream>>>(MKEY,
                                                             n_dst * HEADS);
  init_bias<<<(n_dst * CDIM + 255) / 256, 256, 0, stream>>>(out, bias,
                                                            n_dst * CDIM);

  edge_logits_max<<<((size_t)E * 32 + 255) / 256, 256, 0, stream>>>(
      XL, XR, src, dst, att, EX, MKEY, E);
  edge_exp_den<<<(E * HEADS + 255) / 256, 256, 0, stream>>>(dst, MKEY, EX, DEN,
                                                            E * HEADS);
  edge_scatter<<<((size_t)E * 32 + 255) / 256, 256, 0, stream>>>(
      XL, src, dst, EX, DEN, out, E);
}

extern "C" void kernel_launch(void* const* d_in, const int* in_sizes, int n_in,
                              void* d_out, int out_size, void* d_ws,
                              size_t ws_size, hipStream_t stream) {
  const float* src_emb = (const float*)d_in[0];
  const float* tgt_emb = (const float*)d_in[1];

  // conv param blocks: base = 2 + 6*conv, order Wl,bl,Wr,br,att,bias
  const float* P[4][6];
  for (int c = 0; c < 4; ++c)
    for (int k = 0; k < 6; ++k) P[c][k] = (const float*)d_in[2 + 6 * c + k];

  const float* ln_s_w = (const float*)d_in[26];
  const float* ln_s_b = (const float*)d_in[27];
  const float* ln_t_w = (const float*)d_in[28];
  const float* ln_t_b = (const float*)d_in[29];
  const int* nid_s = (const int*)d_in[30];
  const int* nid_t = (const int*)d_in[31];
  const int* ei_st = (const int*)d_in[32];
  const int* ei_ts = (const int*)d_in[33];
  const int* ei_lbl = (const int*)d_in[34];

  const int n_src = in_sizes[0] / CDIM;   // 50000
  const int n_tgt = in_sizes[1] / CDIM;   // 50000
  const int E     = in_sizes[32] / 2;     // 400000
  const int E_LBL = in_sizes[34] / 2;     // 200000
  const int nmax  = n_src > n_tgt ? n_src : n_tgt;

  // ---- workspace layout (floats) ----
  float* ws = (float*)d_ws;
  size_t off = 0;
  float* XL  = ws + off; off += (size_t)nmax * FDIM;
  float* XR  = ws + off; off += (size_t)nmax * FDIM;
  float* EX  = ws + off; off += (size_t)E * HEADS;
  float* DEN = ws + off; off += (size_t)nmax * HEADS;
  unsigned* MKEY = (unsigned*)(ws + off); off += (size_t)nmax * HEADS;
  float* H_S = ws + off; off += (size_t)n_src * CDIM;
  float* H_T = ws + off; off += (size_t)n_tgt * CDIM;
  float* O_S = ws + off; off += (size_t)n_src * CDIM;
  float* O_T = ws + off; off += (size_t)n_tgt * CDIM;
  (void)ws_size; (void)n_in;

  // ---- layer 1 ----
  // h_t = gatv2(x_s -> x_t, edge_index_st, c1st)
  run_gatv2(src_emb, nid_s, tgt_emb, nid_t, ei_st, E,
            P[0][0], P[0][1], P[0][2], P[0][3], P[0][4], P[0][5],
            n_src, n_tgt, XL, XR, EX, DEN, MKEY, H_T, stream);
  // h_s = gatv2(x_t -> x_s, edge_index_ts, c1ts)
  run_gatv2(tgt_emb, nid_t, src_emb, nid_s, ei_ts, E,
            P[1][0], P[1][1], P[1][2], P[1][3], P[1][4], P[1][5],
            n_tgt, n_src, XL, XR, EX, DEN, MKEY, H_S, stream);

  relu_layernorm<<<(n_src * 32 + 255) / 256, 256, 0, stream>>>(H_S, ln_s_w,
                                                               ln_s_b, n_src);
  relu_layernorm<<<(n_tgt * 32 + 255) / 256, 256, 0, stream>>>(H_T, ln_t_w,
                                                               ln_t_b, n_tgt);

  // ---- layer 2 ----
  // o_t = gatv2(h_s -> h_t, edge_index_st, c2st)
  run_gatv2(H_S, nullptr, H_T, nullptr, ei_st, E,
            P[2][0], P[2][1], P[2][2], P[2][3], P[2][4], P[2][5],
            n_src, n_tgt, XL, XR, EX, DEN, MKEY, O_T, stream);
  // o_s = gatv2(h_t -> h_s, edge_index_ts, c2ts)
  run_gatv2(H_T, nullptr, H_S, nullptr, ei_ts, E,
            P[3][0], P[3][1], P[3][2], P[3][3], P[3][4], P[3][5],
            n_tgt, n_src, XL, XR, EX, DEN, MKEY, O_S, stream);

  // ---- classifier ----
  edge_dot<<<((size_t)E_LBL * 32 + 255) / 256, 256, 0, stream>>>(
      O_S, O_T, ei_lbl, ei_lbl + E_LBL, (float*)d_out, E_LBL);
  (void)out_size;
}